// InfoNCE_8014408974662
// MI455X (gfx1250) — compile-verified
//
#include <hip/hip_runtime.h>
#include <hip/hip_bf16.h>
#include <math.h>

#define N_SAMPLES 512
#define EMB 256
#define HIDDEN 256

typedef __attribute__((ext_vector_type(2)))  float  v2f;
typedef __attribute__((ext_vector_type(4)))  float  v4f;
typedef __attribute__((ext_vector_type(8)))  float  v8f;

// ---------------------------------------------------------------------------
// Phase A: hx = x @ W1[:256], hy = y @ W1[256:]  (both 512x256 @ 256x256, f32)
// One wave32 per 16x16 output tile; K=256 in steps of 4 via V_WMMA_F32_16X16X4_F32.
// Fragment layouts per CDNA5 ISA 7.12.2 (32-bit A 16x4 / C,D 16x16 tables).
// Verified to lower to v_wmma_f32_16x16x4_f32 (8x unrolled) in round 1.
// ---------------------------------------------------------------------------
__global__ __launch_bounds__(32)
void gemm16_kernel(const float* __restrict__ x, const float* __restrict__ y,
                   const float* __restrict__ W1,
                   float* __restrict__ hx, float* __restrict__ hy)
{
    const int tileM = blockIdx.x;          // 0..31  (rows of samples)
    const int tileN = blockIdx.y;          // 0..15  (hidden cols)
    const int which = blockIdx.z;          // 0 -> x@W1x, 1 -> y@W1y
    const float* __restrict__ A = which ? y : x;
    const float* __restrict__ B = W1 + (which ? (EMB * HIDDEN) : 0); // W1 row-major (512,256)
    float* __restrict__ D = which ? hy : hx;

    const int lane = threadIdx.x;          // 0..31
    const int half = lane >> 4;            // 0 | 1
    const int lid  = lane & 15;
    const int row  = tileM * 16 + lid;     // A row owned by this lane
    const int col  = tileN * 16 + lid;     // B col owned by this lane

    v8f acc = {};
    for (int k0 = 0; k0 < EMB; k0 += 4) {
        // A 16x4: lane holds A[row][k0 + 2*half + {0,1}] (merges to global_load_b64)
        v2f af;
        af.x = A[row * EMB + k0 + 2 * half + 0];
        af.y = A[row * EMB + k0 + 2 * half + 1];
        // B 4x16: lane holds B[k0 + 2*half + {0,1}][col]  (row stride HIDDEN)
        v2f bf;
        bf.x = B[(k0 + 2 * half + 0) * HIDDEN + col];
        bf.y = B[(k0 + 2 * half + 1) * HIDDEN + col];
        acc = __builtin_amdgcn_wmma_f32_16x16x4_f32(false, af, false, bf,
                                                    (short)0, acc, false, false);
    }

    // D 16x16 f32: VGPR r -> row = 8*half + r, col = lid
    #pragma unroll
    for (int r = 0; r < 8; ++r) {
        D[(tileM * 16 + half * 8 + r) * HIDDEN + tileN * 16 + lid] = acc[r];
    }
}

// ---------------------------------------------------------------------------
// Phase B: per row i, over all j:
//   t      = sum_k relu(hy[i,k] + hx[j,k] + b1[k]) * W2[k]
//   T1     = softplus(t + b2)                 (T1[i,i] == T0[i])
//   lse[i] = log sum_j exp(T1[i,j])           (online, numerically stable)
// Block = 256 threads (8 waves) per i; wave w handles j = w, w+8, ...
// Lane owns k in [lane*8, lane*8+8) so the 8 hx values per (lane,j) are
// contiguous -> two global_load_b128 per j (fully coalesced across the wave).
// hx is L2-resident (512 KB << 192 MB L2), so streamed re-reads are cheap.
// ---------------------------------------------------------------------------
__global__ __launch_bounds__(256)
void pairwise_kernel(const float* __restrict__ hx, const float* __restrict__ hy,
                     const float* __restrict__ b1, const float* __restrict__ W2,
                     const float* __restrict__ b2,
                     float* __restrict__ lse, float* __restrict__ diag)
{
    const int i    = blockIdx.x;
    const int lane = threadIdx.x & 31;
    const int wave = threadIdx.x >> 5;   // 0..7
    __shared__ float sm[8], ss[8];

    // Per-lane invariants: 8 contiguous k's per lane (k = lane*8 + r)
    const v4f* hy4 = (const v4f*)(hy + i * HIDDEN);
    const v4f* b14 = (const v4f*)b1;
    const v4f* w24 = (const v4f*)W2;
    v4f a0 = hy4[lane * 2 + 0] + b14[lane * 2 + 0];
    v4f a1 = hy4[lane * 2 + 1] + b14[lane * 2 + 1];
    v4f w0 = w24[lane * 2 + 0];
    v4f w1 = w24[lane * 2 + 1];
    const float bias2 = b2[0];

    float m = -1e30f, s = 0.0f;          // online logsumexp state (wave-uniform)
    for (int j = wave; j < N_SAMPLES; j += 8) {
        const v4f* hx4 = (const v4f*)(hx + j * HIDDEN);
        v4f h0 = hx4[lane * 2 + 0];
        v4f h1 = hx4[lane * 2 + 1];
        if (j + 8 < N_SAMPLES)            // prefetch next row for this wave
            __builtin_prefetch(hx + (j + 8) * HIDDEN + lane * 8, 0, 3);

        h0 += a0;  h1 += a1;
        float t = 0.0f;
        #pragma unroll
        for (int e = 0; e < 4; ++e) {
            t = fmaf(fmaxf(h0[e], 0.0f), w0[e], t);
            t = fmaf(fmaxf(h1[e], 0.0f), w1[e], t);
        }
        // butterfly reduction: all 32 lanes end with the full dot product
        #pragma unroll
        for (int off = 16; off > 0; off >>= 1)
            t += __shfl_xor(t, off, 32);

        float z  = t + bias2;
        float T1 = fmaxf(z, 0.0f) + log1pf(__expf(-fabsf(z)));  // stable softplus

        if (j == i && lane == 0) diag[i] = T1;   // T0[i] == T1[i,i]

        if (T1 > m) { s = s * __expf(m - T1) + 1.0f; m = T1; }  // wave-uniform branch
        else        { s += __expf(T1 - m); }
    }

    if (lane == 0) { sm[wave] = m; ss[wave] = s; }
    __syncthreads();
    if (threadIdx.x == 0) {
        float M = sm[0];
        #pragma unroll
        for (int wv = 1; wv < 8; ++wv) M = fmaxf(M, sm[wv]);
        float S = 0.0f;
        #pragma unroll
        for (int wv = 0; wv < 8; ++wv) S += ss[wv] * __expf(sm[wv] - M);
        lse[i] = M + __logf(S);
    }
}

// ---------------------------------------------------------------------------
// Phase C: loss = mean(lse) - log(N) - mean(T0)
// ---------------------------------------------------------------------------
__global__ __launch_bounds__(256)
void finalize_kernel(const float* __restrict__ lse, const float* __restrict__ diag,
                     float* __restrict__ out)
{
    __shared__ float sl[256], sd[256];
    const int t = threadIdx.x;
    sl[t] = lse[t] + lse[t + 256];
    sd[t] = diag[t] + diag[t + 256];
    __syncthreads();
    for (int off = 128; off > 0; off >>= 1) {
        if (t < off) { sl[t] += sl[t + off]; sd[t] += sd[t + off]; }
        __syncthreads();
    }
    if (t == 0) {
        const float invN = 1.0f / (float)N_SAMPLES;
        out[0] = sl[0] * invN - logf((float)N_SAMPLES) - sd[0] * invN;
    }
}

extern "C" void kernel_launch(void* const* d_in, const int* in_sizes, int n_in,
                              void* d_out, int out_size, void* d_ws, size_t ws_size,
                              hipStream_t stream)
{
    const float* x  = (const float*)d_in[0];  // (512, 256)
    const float* y  = (const float*)d_in[1];  // (512, 256)
    const float* W1 = (const float*)d_in[2];  // (512, 256)
    const float* b1 = (const float*)d_in[3];  // (256,)
    const float* W2 = (const float*)d_in[4];  // (256, 1)
    const float* b2 = (const float*)d_in[5];  // (1,)

    float* hx   = (float*)d_ws;                       // 512*256
    float* hy   = hx + N_SAMPLES * HIDDEN;            // 512*256
    float* lse  = hy + N_SAMPLES * HIDDEN;            // 512
    float* diag = lse + N_SAMPLES;                    // 512

    gemm16_kernel<<<dim3(N_SAMPLES / 16, HIDDEN / 16, 2), 32, 0, stream>>>(x, y, W1, hx, hy);
    pairwise_kernel<<<N_SAMPLES, 256, 0, stream>>>(hx, hy, b1, W2, b2, lse, diag);
    finalize_kernel<<<1, 256, 0, stream>>>(lse, diag, (float*)d_out);
}